// SCAN_31851477467669
// MI455X (gfx1250) — compile-verified
//
#include <hip/hip_runtime.h>
#include <math.h>
#include <stdint.h>

#define I_N 128
#define C_N 64
#define R_N 36
#define L_N 32
#define D_N 512
#define CGRP 8   // captions per workgroup (image stays LDS-resident)

#define IS 516   // imgs LDS row stride (floats): 516*4 B -> 16B aligned rows, conflict-free frags
#define CS 516   // caps LDS row stride
#define AS 33    // attnA (r-major) stride
#define TS 40    // attnT (l-major, K-padded to 40) stride
#define WS 132   // wctx chunk stride
#define CHUNK 128

#define EPSF   1e-8f
#define LLSE   6.0f
#define LSM    9.0f
#define SLOPE  0.1f

typedef __attribute__((ext_vector_type(2))) float v2f;
typedef __attribute__((ext_vector_type(8))) float v8f;

__device__ __forceinline__ float wave_sum(float v) {
#pragma unroll
  for (int o = 16; o; o >>= 1) v += __shfl_xor(v, o, 32);
  return v;
}
__device__ __forceinline__ float wave_max(float v) {
#pragma unroll
  for (int o = 16; o; o >>= 1) v = fmaxf(v, __shfl_xor(v, o, 32));
  return v;
}

// CDNA5 async global->LDS copy, 16B per lane, tracked by ASYNCcnt.
// LDS destination operand = per-lane LDS byte offset (low 32 bits of the
// generic pointer: shared aperture occupies addr[63:32]).
__device__ __forceinline__ void async_copy_b128(uint32_t lds_byte_off, const void* gaddr) {
  asm volatile("global_load_async_to_lds_b128 %0, %1, off"
               :: "v"(lds_byte_off), "v"(gaddr) : "memory");
}
__device__ __forceinline__ void wait_async0() {
  asm volatile("s_wait_asynccnt 0" ::: "memory");
}

__global__ __launch_bounds__(192)
void scan_t2i_kernel(const float* __restrict__ images,
                     const float* __restrict__ captions,
                     const long long* __restrict__ cap_lens,
                     float* __restrict__ out)
{
  // Total static LDS: 162,752 B <= 160 KB -> two workgroups co-reside per
  // 320 KB WGP (12 resident waves) to hide barrier/epilogue bubbles.
  __shared__ float imgs[R_N * IS];     //  74,304 B (resident across all 8 captions)
  __shared__ float caps[L_N * CS];     //  66,048 B (restaged per caption, async)
  __shared__ float attnT[L_N * TS];    //   5,120 B
  __shared__ float wctx[L_N * WS];     //  16,896 B  (aliased as attnA early on)
  __shared__ float num_l[L_N], nn_l[L_N], cc_l[L_N];   // 384 B

  // attnA (36x33 = 4,752 B) lives only between GEMM1 and the softmax
  // transpose; wctx is written strictly after that barrier -> safe alias.
  float* const attnA = wctx;

  const int ig   = blockIdx.x >> 3;    // image index 0..127
  const int cg   = blockIdx.x & 7;     // caption group 0..7
  const int tid  = threadIdx.x;
  const int wave = tid >> 5;
  const int lane = tid & 31;
  const int half = lane >> 4;          // 0 -> K,K+1 ; 1 -> K+2,K+3
  const int lx   = lane & 15;

  // ---------- stage images[ig] (36x512) once; LDS-resident for 8 captions ----------
  {
    const float4* gi = (const float4*)(images + (size_t)ig * R_N * D_N);
    for (int idx = tid; idx < R_N * (D_N / 4); idx += 192) {
      int r = idx >> 7, c4 = idx & 127;
      *(float4*)&imgs[r * IS + c4 * 4] = gi[idx];
    }
  }

  for (int j = 0; j < CGRP; ++j) {
    const int bc = cg * CGRP + j;
    const int caplen = (int)cap_lens[bc];

    // ---------- async-stage captions[bc] (32x512) straight into LDS ----------
    {
      const float4* gc = (const float4*)(captions + (size_t)bc * L_N * D_N);
      for (int idx = tid; idx < L_N * (D_N / 4); idx += 192) {
        int r = idx >> 7, c4 = idx & 127;
        async_copy_b128((uint32_t)(uintptr_t)&caps[r * CS + c4 * 4], gc + idx);
      }
      wait_async0();
    }
    __syncthreads();   // imgs (j==0) + caps visible to all waves

    // ---------- GEMM1: attn(36x32) = imgs(36x512) . capsT(512x32) ----------
    // 6 tiles of 16x16, one per wave; K = 512 in steps of 4.
    // M-tiles at rows 0, 16, 20 (third tile OVERLAPS rows 20..35 so every
    // A-row is valid -> no zero-masking VALU in the hot loop; the epilogue
    // discards the recomputed rows 20..31 of tile 2).
    {
      const int mt = wave >> 1;                 // 0..2
      const int nt = wave & 1;                  // 0..1
      const int mbase = (mt == 2) ? 20 : mt * 16;
      const int mrow = (mbase + lx) * IS;       // A row (M) for this lane
      const int koff = half * 2;
      const int nrow = (nt * 16 + lx) * CS;     // B col (N = l) for this lane
      v8f acc = {};
      for (int k = 0; k < D_N; k += 4) {
        float2 av = *(const float2*)&imgs[mrow + k + koff];
        float2 bv = *(const float2*)&caps[nrow + k + koff];
        v2f a; a[0] = av.x; a[1] = av.y;
        v2f b; b[0] = bv.x; b[1] = bv.y;
        acc = __builtin_amdgcn_wmma_f32_16x16x4_f32(false, a, false, b,
                                                    (short)0, acc, false, false);
      }
      // fused epilogue: leaky_relu + caption-length mask, store r-major
      const int l = nt * 16 + lx;
      const float lmask = (l < caplen) ? 1.0f : 0.0f;
#pragma unroll
      for (int p = 0; p < 8; ++p) {
        int r = mbase + half * 8 + p;          // C/D layout: lanes16-31 hold M+8
        if (mt < 2 || r >= 32) {               // tile 2 keeps only rows 32..35
          float v = acc[p];
          v = (v > 0.0f) ? v : (SLOPE * v);
          attnA[r * AS + l] = v * lmask;
        }
      }
    }
    __syncthreads();

    // ---------- L2-normalize each row r over l ----------
    if (tid < R_N) {
      float s = 0.0f;
      for (int l = 0; l < L_N; ++l) { float v = attnA[tid * AS + l]; s += v * v; }
      float inv = 1.0f / (sqrtf(s) + EPSF);
      for (int l = 0; l < L_N; ++l) attnA[tid * AS + l] *= inv;
    }
    __syncthreads();

    // ---------- softmax(9*attn) over r per column l; write transposed ----------
    if (tid < L_N) {
      float mx = -1e30f;
      for (int r = 0; r < R_N; ++r) mx = fmaxf(mx, attnA[r * AS + tid]);
      float s = 0.0f;
      for (int r = 0; r < R_N; ++r) {
        float e = expf(LSM * (attnA[r * AS + tid] - mx));
        attnT[tid * TS + r] = e;
        s += e;
      }
      float inv = 1.0f / s;
      for (int r = 0; r < R_N; ++r) attnT[tid * TS + r] *= inv;
#pragma unroll
      for (int r = R_N; r < TS; ++r) attnT[tid * TS + r] = 0.0f;  // K-pad to 40
      num_l[tid] = 0.0f; nn_l[tid] = 0.0f; cc_l[tid] = 0.0f;
    }
    __syncthreads();   // attnA region now dead -> wctx may reuse it

    // ---------- GEMM2 (chunked over d) + fused cosine reductions ----------
    // wctx(32x512) = attnT(32x36) . imgs(36x512); 128-wide d-chunks, K=36 -> 9 steps
    for (int d0 = 0; d0 < D_N; d0 += CHUNK) {
      for (int t = wave; t < 16; t += 6) {       // 2 l-tiles x 8 d-tiles per chunk
        const int lt = t >> 3, dt = t & 7;
        const int m  = lt * 16 + lx;             // A row (M = l)
        const int koff = half * 2;
        const int dcol = d0 + dt * 16 + lx;      // B col (N = d)
        v8f acc = {};
#pragma unroll
        for (int ks = 0; ks < 9; ++ks) {
          int kk = ks * 4 + koff;                // K = r, exactly 36
          float2 av = *(const float2*)&attnT[m * TS + kk];
          v2f a; a[0] = av.x; a[1] = av.y;
          v2f b; b[0] = imgs[kk * IS + dcol]; b[1] = imgs[(kk + 1) * IS + dcol];
          acc = __builtin_amdgcn_wmma_f32_16x16x4_f32(false, a, false, b,
                                                      (short)0, acc, false, false);
        }
#pragma unroll
        for (int p = 0; p < 8; ++p) {
          int l = lt * 16 + half * 8 + p;
          wctx[l * WS + dt * 16 + lx] = acc[p];
        }
      }
      __syncthreads();

      for (int l = wave; l < L_N; l += 6) {      // each l owned by one wave
        float pn = 0.0f, pw = 0.0f, pc = 0.0f;
        for (int cix = lane; cix < CHUNK; cix += 32) {
          float w  = wctx[l * WS + cix];
          float cp = caps[l * CS + d0 + cix];
          pn += cp * w;
          pw += w * w;
          pc += cp * cp;
        }
        pn = wave_sum(pn); pw = wave_sum(pw); pc = wave_sum(pc);
        if (lane == 0) { num_l[l] += pn; nn_l[l] += pw; cc_l[l] += pc; }
      }
      __syncthreads();
    }

    // ---------- masked LogSumExp over l, write sims[i][c] ----------
    if (wave == 0) {
      const int l = lane;
      float nw = fmaxf(sqrtf(nn_l[l]), EPSF);
      float nc = fmaxf(sqrtf(cc_l[l]), EPSF);
      float rs = num_l[l] / (nc * nw);
      float sc = (l < caplen) ? rs * LLSE : -INFINITY;
      float mx = wave_max(sc);
      float s  = wave_sum(expf(sc - mx));
      if (lane == 0) out[ig * C_N + bc] = (mx + logf(s)) / LLSE;
    }
    // no barrier needed here: next iteration's caption staging is fenced by the
    // post-staging __syncthreads before anyone reads caps; num/nn/cc are only
    // rewritten after further barriers that wave 0 participates in.
  }
}

extern "C" void kernel_launch(void* const* d_in, const int* in_sizes, int n_in,
                              void* d_out, int out_size, void* d_ws, size_t ws_size,
                              hipStream_t stream) {
  const float*     images   = (const float*)d_in[0];
  const float*     captions = (const float*)d_in[1];
  const long long* cap_lens = (const long long*)d_in[2];   // int64 per reference
  float* out = (float*)d_out;

  dim3 grid(I_N * (C_N / CGRP));   // 1024 workgroups: (image, caption-group-of-8)
  dim3 block(192);                 // 6 wave32 waves
  hipLaunchKernelGGL(scan_t2i_kernel, grid, block, 0, stream,
                     images, captions, cap_lens, out);
}